// MyGRU_67216238182881
// MI455X (gfx1250) — compile-verified
//
#include <hip/hip_runtime.h>

// Problem constants (from reference)
#define kT   4096
#define kB   64
#define kI   256
#define kH   256
#define kNWG 16     // persistent workgroups; each owns 16 hidden units
#define kJW  16
#define LDA  260    // padded LDS row stride (floats): 16 rows x 4 cols hit all 64 banks

typedef __attribute__((ext_vector_type(2))) float v2f;
typedef __attribute__((ext_vector_type(8))) float v8f;
typedef int async_v4i __attribute__((vector_size(16)));   // exact builtin param elem type

// LDS layout (in floats): weights + double-buffered x + h + staging + bias
#define WIH_OFF  0
#define WHH_OFF  (WIH_OFF + 48 * LDA)
#define X0_OFF   (WHH_OFF + 48 * LDA)
#define X1_OFF   (X0_OFF + 64 * LDA)
#define H_OFF    (X1_OFF + 64 * LDA)
#define STG_OFF  (H_OFF + 64 * LDA)
#define BIAS_OFF (STG_OFF + 4 * 64 * 16)
#define SMEM_FLOATS (BIAS_OFF + 4 * 16)
#define SMEM_BYTES  (SMEM_FLOATS * 4)   // 316,160 B  (< 320KB WGP limit)

#if defined(__gfx1250__) && __has_builtin(__builtin_amdgcn_global_load_async_to_lds_b128)
#define HAVE_ASYNC 1
#else
#define HAVE_ASYNC 0
#endif

#if defined(__gfx1250__) && __has_builtin(__builtin_amdgcn_s_wait_asynccnt)
#define WAIT_ASYNC(n) __builtin_amdgcn_s_wait_asynccnt(n)
#elif defined(__gfx1250__)
#define WAIT_ASYNC(n) asm volatile("s_wait_asynccnt " #n)
#else
#define WAIT_ASYNC(n)
#endif

#if HAVE_ASYNC
// Async-DMA one 64KB [64][256] row-major block into padded [64][LDA] LDS region.
// Each thread issues 16 b128 transfers; wave ASYNCcnt += 16.
__device__ __forceinline__ void async_copy_block(const float* __restrict__ g,
                                                 unsigned ldsByteBase, int tid) {
    #pragma unroll
    for (int it = 0; it < 16; ++it) {
        const int i   = tid + it * 256;       // float4 chunk id, 0..4095
        const int row = i >> 6;
        const int c4  = i & 63;
        const unsigned dst = ldsByteBase + (unsigned)((row * LDA + c4 * 4) * 4);
        __builtin_amdgcn_global_load_async_to_lds_b128(
            (__attribute__((address_space(1))) async_v4i*)(unsigned long long)(g + (size_t)i * 4),
            (__attribute__((address_space(3))) async_v4i*)dst,
            0, 0);
    }
}
#endif

__device__ __forceinline__ v2f lds2(const float* p) { v2f v; v.x = p[0]; v.y = p[1]; return v; }

// Fused K-loop: 3 independent WMMA accumulator chains.
//   acc0a: x-side of (r|z), acc0b: h-side of (r|z), accN: nx (P=0, A=x) or nh (P=1, A=h)
template <bool P>
__device__ __forceinline__ void gemm_kloop(float* smem, int offAX, int offAH,
                                           int offRZx, int offRZh, int offN, int hi,
                                           v8f& acc0a, v8f& acc0b, v8f& accN) {
    #pragma unroll 4
    for (int k = 0; k < kI; k += 4) {
        const int k2 = k + 2 * hi;
        v2f ax = lds2(smem + offAX  + k2);
        v2f ah = lds2(smem + offAH  + k2);
        v2f bx = lds2(smem + offRZx + k2);
        v2f bh = lds2(smem + offRZh + k2);
        v2f bn = lds2(smem + offN   + k2);
        acc0a = __builtin_amdgcn_wmma_f32_16x16x4_f32(false, ax, false, bx, (short)0, acc0a, false, false);
        acc0b = __builtin_amdgcn_wmma_f32_16x16x4_f32(false, ah, false, bh, (short)0, acc0b, false, false);
        if (P)
            accN = __builtin_amdgcn_wmma_f32_16x16x4_f32(false, ah, false, bn, (short)0, accN, false, false);
        else
            accN = __builtin_amdgcn_wmma_f32_16x16x4_f32(false, ax, false, bn, (short)0, accN, false, false);
    }
}

__global__ __launch_bounds__(256, 1)
void gru_persistent(const float* __restrict__ inp,
                    const float* __restrict__ h0,
                    const float* __restrict__ w_ih,
                    const float* __restrict__ b_ih,
                    const float* __restrict__ w_hh,
                    const float* __restrict__ b_hh,
                    float* __restrict__ out,
                    unsigned int* __restrict__ bar)
{
    extern __shared__ float smem[];
    float* sStg  = smem + STG_OFF;   // [4][64][16] kinds: 0=r,1=z,2=nx,3=nh
    float* sBias = smem + BIAS_OFF;  // [4][16]

    const int wg  = blockIdx.x;
    const int tid = threadIdx.x;
    const int j0  = wg * kJW;

    // ---- one-time preload: weight slices + fused biases ----
    for (int i = tid; i < 48 * kI; i += 256) {
        const int row  = i >> 8;          // 0..47  (0-15=r, 16-31=z, 32-47=n)
        const int col  = i & 255;
        const int g    = row >> 4;        // gate 0=r,1=z,2=n
        const int j    = row & 15;
        const int grow = g * kH + j0 + j; // global gate row
        smem[WIH_OFF + row * LDA + col] = w_ih[grow * kI + col];
        smem[WHH_OFF + row * LDA + col] = w_hh[grow * kH + col];
    }
    if (tid < kJW) {
        sBias[0 * 16 + tid] = b_ih[j0 + tid]          + b_hh[j0 + tid];          // r
        sBias[1 * 16 + tid] = b_ih[kH + j0 + tid]     + b_hh[kH + j0 + tid];     // z
        sBias[2 * 16 + tid] = b_ih[2 * kH + j0 + tid];                           // nx
        sBias[3 * 16 + tid] = b_hh[2 * kH + j0 + tid];                           // nh
    }

#if HAVE_ASYNC
    const unsigned ldsBase = (unsigned)(unsigned long long)smem;  // LDS byte offset of dyn-shared base
    // Prefetch x_0 into buffer 0 while we finish setup.
    async_copy_block(inp, ldsBase + X0_OFF * 4, tid);
#endif
    __syncthreads();

    const int wave = tid >> 5;
    const int lane = tid & 31;
    const int m    = wave >> 1;       // batch tile 0..3 (rows m*16..m*16+15)
    const int p    = wave & 1;        // 0: {r, nx}   1: {z, nh}
    const int m0   = m * 16;
    const int nl   = lane & 15;
    const int hi   = lane >> 4;

    const int arow   = (m0 + nl) * LDA;            // A-fragment row (within X/H block)
    const int wrowRZ = ((p ? 16 : 0) + nl) * LDA;  // B rows for r or z
    const int wrowN  = (32 + nl) * LDA;            // B rows for n

    const int offAH  = H_OFF + arow;
    const int offRZx = WIH_OFF + wrowRZ;
    const int offRZh = WHH_OFF + wrowRZ;
    const int offN   = (p ? WHH_OFF : WIH_OFF) + wrowN;

    const float bias0 = sBias[(p ? 1 : 0) * 16 + nl];
    const float bias1 = sBias[(p ? 3 : 2) * 16 + nl];

    for (int t = 0; t < kT; ++t) {
        const int xbase = (t & 1) ? X1_OFF : X0_OFF;
        const float* hsrc = (t == 0) ? h0 : (out + (size_t)(t - 1) * (kB * kH));

#if HAVE_ASYNC
        // h_t is on the critical path: async-DMA it now; then launch the x_{t+1}
        // prefetch into the other buffer; wait for h (and x_t) with the prefetch
        // still in flight (async loads complete in order per wave).
        async_copy_block(hsrc, ldsBase + H_OFF * 4, tid);
        if (t + 1 < kT) {
            const int xnext = (t & 1) ? X0_OFF : X1_OFF;
            async_copy_block(inp + (size_t)(t + 1) * (kB * kI), ldsBase + xnext * 4, tid);
            WAIT_ASYNC(16);
        } else {
            WAIT_ASYNC(0);
        }
#else
        // Fallback: synchronous coalesced staging of x_t and h_{t-1}.
        const float* xsrc = inp + (size_t)t * (kB * kI);
        for (int i = tid; i < (kB * kI) / 4; i += 256) {
            const int row = i >> 6;
            const int c4  = i & 63;
            float4 xv = ((const float4*)xsrc)[i];
            float4 hv = ((const float4*)hsrc)[i];
            *((float4*)(smem + xbase + row * LDA) + c4) = xv;
            *((float4*)(smem + H_OFF + row * LDA) + c4) = hv;
        }
#endif
        __syncthreads();

        v8f acc0a, acc0b, accN;
        #pragma unroll
        for (int r = 0; r < 8; ++r) { acc0a[r] = bias0; acc0b[r] = 0.0f; accN[r] = bias1; }

        const int offAX = xbase + arow;
        if (p)
            gemm_kloop<true >(smem, offAX, offAH, offRZx, offRZh, offN, hi, acc0a, acc0b, accN);
        else
            gemm_kloop<false>(smem, offAX, offAH, offRZx, offRZh, offN, hi, acc0a, acc0b, accN);

        // ---- stage accumulator tiles to LDS (C/D layout: VGPR r -> M=r+8*hi) ----
        {
            const int kind0 = p;       // 0=r or 1=z
            const int kind1 = 2 + p;   // 2=nx or 3=nh
            #pragma unroll
            for (int r = 0; r < 8; ++r) {
                const int row = m0 + r + 8 * hi;
                sStg[kind0 * (64 * 16) + row * 16 + nl] = acc0a[r] + acc0b[r];
                sStg[kind1 * (64 * 16) + row * 16 + nl] = accN[r];
            }
        }
        __syncthreads();

        // ---- elementwise gate math: h_new = (1-z)*tanh(nx + r*nh) + z*h ----
        float* orow = out + (size_t)t * (kB * kH);
        #pragma unroll
        for (int it = 0; it < 4; ++it) {
            const int i   = tid + it * 256;  // 0..1023 over [64 rows][16 j]
            const int row = i >> 4;
            const int j   = i & 15;
            const float rv = sStg[0 * 1024 + i];
            const float zv = sStg[1 * 1024 + i];
            const float nx = sStg[2 * 1024 + i];
            const float nh = sStg[3 * 1024 + i];
            const float rg = 1.0f / (1.0f + __expf(-rv));
            const float zg = 1.0f / (1.0f + __expf(-zv));
            const float ng = tanhf(nx + rg * nh);
            const float hp = smem[H_OFF + row * LDA + j0 + j];
            const float hn = (1.0f - zg) * ng + zg * hp;
            orow[row * kH + j0 + j] = hn;
            if (t == kT - 1)
                out[(size_t)kT * (kB * kH) + row * kH + j0 + j] = hn;  // h_final
        }

        // ---- device-wide step barrier (release add / acquire spin) ----
        __threadfence();
        __syncthreads();
        if (tid == 0) {
            __hip_atomic_fetch_add(bar, 1u, __ATOMIC_RELEASE, __HIP_MEMORY_SCOPE_AGENT);
            const unsigned target = (unsigned)(t + 1) * kNWG;
            while (__hip_atomic_load(bar, __ATOMIC_ACQUIRE, __HIP_MEMORY_SCOPE_AGENT) < target) {
                __builtin_amdgcn_s_sleep(2);
            }
        }
        __syncthreads();
        __threadfence();  // every wave: invalidate caches before re-reading h from L2
    }
}

extern "C" void kernel_launch(void* const* d_in, const int* in_sizes, int n_in,
                              void* d_out, int out_size, void* d_ws, size_t ws_size,
                              hipStream_t stream) {
    (void)in_sizes; (void)n_in; (void)out_size; (void)ws_size;
    const float* inp  = (const float*)d_in[0];
    const float* h0   = (const float*)d_in[1];
    const float* w_ih = (const float*)d_in[2];
    const float* b_ih = (const float*)d_in[3];
    const float* w_hh = (const float*)d_in[4];
    const float* b_hh = (const float*)d_in[5];
    float*        out = (float*)d_out;
    unsigned int* bar = (unsigned int*)d_ws;

    // Allow ~309KB dynamic LDS (CDNA5: up to 320KB per workgroup).
    (void)hipFuncSetAttribute(reinterpret_cast<const void*>(gru_persistent),
                              hipFuncAttributeMaxDynamicSharedMemorySize, SMEM_BYTES);

    // Zero the grid-barrier counter (deterministic per call; capture-legal).
    (void)hipMemsetAsync(d_ws, 0, 64, stream);

    gru_persistent<<<kNWG, 256, SMEM_BYTES, stream>>>(inp, h0, w_ih, b_ih, w_hh, b_hh, out, bar);
}